// HeteroGNNFraudDetector_20323785244837
// MI455X (gfx1250) — compile-verified
//
#include <hip/hip_runtime.h>

// ---------------------------------------------------------------------------
// HeteroGNN fraud detector on MI455X (gfx1250, wave32).
// f32 end-to-end; node transforms use V_WMMA_F32_16X16X4_F32 with pre-swizzled
// float2 LDS staging of B (one ds_load_b64 per fragment); edge scatter uses
// global f32 atomics (L2-resident accumulators) + next-row prefetch.
// ---------------------------------------------------------------------------

#define NU 100000
#define NT 200000
#define NE 1000000
#define HD 64
#define NLAYERS 3

typedef __attribute__((ext_vector_type(2))) float v2f;
typedef __attribute__((ext_vector_type(8))) float v8f;

static __device__ __forceinline__ v8f wmma4(v2f a, v2f b, v8f c) {
  // 8-arg form: (neg_a, A, neg_b, B, c_mod, C, reuse_a, reuse_b)
  return __builtin_amdgcn_wmma_f32_16x16x4_f32(false, a, false, b, (short)0, c,
                                               false, false);
}

// C[M x 64] = A[M x K] @ B[K x 64] (+ bias, optional).  M % 16 == 0, K % 4 == 0.
// 256 threads = 8 waves; wave w computes rows [ (blk*8+w)*16 , +16 ) x all 64 cols.
// B staged in LDS pre-paired: Bs2[(k/2)*64 + n] = {B[k][n], B[k+1][n]} so each
// WMMA B fragment is a single aligned ds_load_b64.
__global__ void gemm_n64(const float* __restrict__ A, const float* __restrict__ B,
                         const float* __restrict__ bias, float* __restrict__ C,
                         int M, int K) {
  __shared__ float2 Bs2[32 * 64];  // up to K=64 -> 32 row-pairs x 64 cols = 16 KB
  const int npair = (K >> 1) * 64;
  for (int i = threadIdx.x; i < npair; i += blockDim.x) {
    const int k2 = i >> 6;        // row-pair index
    const int n = i & 63;         // column
    Bs2[i] = make_float2(B[(2 * k2) * 64 + n], B[(2 * k2 + 1) * 64 + n]);
  }
  __syncthreads();

  const int wave = threadIdx.x >> 5;
  const int lane = threadIdx.x & 31;
  const int hi = lane >> 4;   // 0 or 1
  const int lo = lane & 15;
  const int rowTile = blockIdx.x * 8 + wave;
  if (rowTile * 16 >= M) return;  // whole wave exits -> EXEC all-1s for WMMA

  const float* Arow = A + (size_t)(rowTile * 16 + lo) * K;
  v8f c0 = {}, c1 = {}, c2 = {}, c3 = {};
  for (int k0 = 0; k0 < K; k0 += 4) {
    const int ka = k0 + 2 * hi;                 // even
    const v2f a = *(const v2f*)(Arow + ka);     // global_load_b64
    const v2f* bp = (const v2f*)&Bs2[(ka >> 1) * 64 + lo];
    const v2f b0 = bp[0];    // ds_load_b64 each
    const v2f b1 = bp[16];
    const v2f b2 = bp[32];
    const v2f b3 = bp[48];
    c0 = wmma4(a, b0, c0);
    c1 = wmma4(a, b1, c1);
    c2 = wmma4(a, b2, c2);
    c3 = wmma4(a, b3, c3);
  }
  const float bb0 = bias ? bias[lo]      : 0.f;
  const float bb1 = bias ? bias[lo + 16] : 0.f;
  const float bb2 = bias ? bias[lo + 32] : 0.f;
  const float bb3 = bias ? bias[lo + 48] : 0.f;
#pragma unroll
  for (int r = 0; r < 8; ++r) {
    float* crow = C + (size_t)(rowTile * 16 + 8 * hi + r) * 64;
    crow[lo]      = c0[r] + bb0;
    crow[lo + 16] = c1[r] + bb1;
    crow[lo + 32] = c2[r] + bb2;
    crow[lo + 48] = c3[r] + bb3;
  }
}

__global__ void fill_f32(float* __restrict__ p, float v, long n) {
  long i = (long)blockIdx.x * blockDim.x + threadIdx.x;
  const long st = (long)gridDim.x * blockDim.x;
  for (; i < n; i += st) p[i] = v;
}

__global__ void count_deg(const int* __restrict__ idx, float* __restrict__ deg, int n) {
  int i = blockIdx.x * blockDim.x + threadIdx.x;
  const int st = gridDim.x * blockDim.x;
  for (; i < n; i += st) atomicAdd(deg + idx[i], 1.0f);
}

__global__ void deg_inv_self(float* __restrict__ d, int n) {  // d := rsqrt(d+1)
  int i = blockIdx.x * blockDim.x + threadIdx.x;
  const int st = gridDim.x * blockDim.x;
  for (; i < n; i += st) d[i] = rsqrtf(d[i] + 1.0f);
}

__global__ void deg_inv_bip(float* __restrict__ d, int n) {  // d := d>0 ? rsqrt(d) : 0
  int i = blockIdx.x * blockDim.x + threadIdx.x;
  const int st = gridDim.x * blockDim.x;
  for (; i < n; i += st) {
    float v = d[i];
    d[i] = (v > 0.f) ? rsqrtf(v) : 0.f;
  }
}

// acc[i,:] += m[i,:] * dinv[i]^2   (self-loop contribution)
__global__ void add_self(float* __restrict__ acc, const float* __restrict__ m,
                         const float* __restrict__ dinv, long n64) {
  long i = (long)blockIdx.x * blockDim.x + threadIdx.x;
  const long st = (long)gridDim.x * blockDim.x;
  for (; i < n64; i += st) {
    const float s = dinv[i >> 6];
    acc[i] += m[i] * s * s;
  }
}

// One wave per edge: each lane handles 2 of the 64 channels.
// Prefetch the next edge's gather row to overlap HBM latency with atomics.
__global__ void scatter_edges(const float* __restrict__ m, float* __restrict__ acc,
                              const int* __restrict__ src, const int* __restrict__ dst,
                              const float* __restrict__ dinv_s,
                              const float* __restrict__ dinv_d, int nE) {
  const int lane = threadIdx.x & 31;
  int w = (blockIdx.x * blockDim.x + threadIdx.x) >> 5;
  const int nw = (gridDim.x * blockDim.x) >> 5;
  for (int e = w; e < nE; e += nw) {
    const int s = src[e];
    const int d = dst[e];
    const int en = e + nw;
    if (en < nE) {  // global_prefetch_b8 of next source row (lane-strided)
      __builtin_prefetch(m + (size_t)src[en] * 64 + lane * 2, 0, 1);
    }
    const float nrm = dinv_s[s] * dinv_d[d];
    const float2 v = ((const float2*)(m + (size_t)s * 64))[lane];
    float* ar = acc + (size_t)d * 64 + lane * 2;
    atomicAdd(ar, v.x * nrm);
    atomicAdd(ar + 1, v.y * nrm);
  }
}

// h = relu(0.5*(acc + ba + bb))
__global__ void combine_relu(float* __restrict__ h, const float* __restrict__ acc,
                             const float* __restrict__ ba, const float* __restrict__ bb,
                             long n64) {
  long i = (long)blockIdx.x * blockDim.x + threadIdx.x;
  const long st = (long)gridDim.x * blockDim.x;
  for (; i < n64; i += st) {
    const int c = (int)(i & 63);
    const float v = 0.5f * (acc[i] + ba[c] + bb[c]);
    h[i] = v > 0.f ? v : 0.f;
  }
}

__global__ void colsum(const float* __restrict__ h, float* __restrict__ out, int rows) {
  const int tid = blockIdx.x * blockDim.x + threadIdx.x;
  const int c = tid & 63;
  const int rst = (gridDim.x * blockDim.x) >> 6;
  float s = 0.f;
  for (int r = tid >> 6; r < rows; r += rst) s += h[(size_t)r * 64 + c];
  atomicAdd(out + c, s);
}

__global__ void final_mlp(const float* __restrict__ su, const float* __restrict__ st,
                          const float* __restrict__ W1, const float* __restrict__ b1,
                          const float* __restrict__ W2, const float* __restrict__ b2,
                          float* __restrict__ out) {
  __shared__ float g[64];
  __shared__ float h1[64];
  const int t = threadIdx.x;  // 64 threads
  g[t] = 0.5f * (su[t] * (1.0f / NU) + st[t] * (1.0f / NT));
  __syncthreads();
  float a = b1[t];
  for (int c = 0; c < 64; ++c) a += g[c] * W1[c * 64 + t];
  h1[t] = a > 0.f ? a : 0.f;
  __syncthreads();
  if (t == 0) {
    float acc = b2[0];
    for (int j = 0; j < 64; ++j) acc += h1[j] * W2[j];
    out[0] = 1.0f / (1.0f + __expf(-acc));
  }
}

extern "C" void kernel_launch(void* const* d_in, const int* in_sizes, int n_in,
                              void* d_out, int out_size, void* d_ws, size_t ws_size,
                              hipStream_t stream) {
  (void)in_sizes; (void)n_in; (void)out_size; (void)ws_size;
  const float* x_user  = (const float*)d_in[0];
  const float* x_txn   = (const float*)d_in[1];
  const int*   ei_u2u  = (const int*)d_in[2];
  const int*   ei_t2t  = (const int*)d_in[3];
  const int*   ei_u2t  = (const int*)d_in[4];
  const int*   ei_t2u  = (const int*)d_in[5];
  const float* W_emb_u = (const float*)d_in[6];
  const float* b_emb_u = (const float*)d_in[7];
  const float* W_emb_t = (const float*)d_in[8];
  const float* b_emb_t = (const float*)d_in[9];
  const float* convW   = (const float*)d_in[10];
  const float* convB   = (const float*)d_in[11];
  const float* W1      = (const float*)d_in[12];
  const float* b1      = (const float*)d_in[13];
  const float* W2      = (const float*)d_in[14];
  const float* b2      = (const float*)d_in[15];
  float* out = (float*)d_out;

  // --- workspace carve-up (floats) ---
  float* ws = (float*)d_ws;
  size_t off = 0;
  auto carve = [&](size_t n) { float* p = ws + off; off += n; return p; };
  float* hu      = carve((size_t)NU * HD);
  float* ht      = carve((size_t)NT * HD);
  float* m_a     = carve((size_t)NU * HD);   // transformed user features
  float* m_b     = carve((size_t)NT * HD);   // transformed txn features
  float* acc_u   = carve((size_t)NU * HD);   // } contiguous: zeroed together
  float* acc_t   = carve((size_t)NT * HD);   // }
  float* dinv_u  = carve(NU);                // } contiguous degree block + gsum
  float* dinv_t  = carve(NT);
  float* dsi_u2t = carve(NU);
  float* ddi_u2t = carve(NT);
  float* dsi_t2u = carve(NT);
  float* ddi_t2u = carve(NU);
  float* gsum    = carve(128);               // [0:64]=user colsum, [64:128]=txn

  const int B256 = 256;
  const int gBig = 2048;   // grid-stride kernels
  const int gCnt = 2048;
  const int gemmBlkU = (NU / 16 + 7) / 8;    // 782
  const int gemmBlkT = (NT / 16 + 7) / 8;    // 1563

  // --- degree normalization (edge structure is layer-invariant) ---
  fill_f32<<<gBig, B256, 0, stream>>>(dinv_u, 0.f, (long)3 * NU + 3L * NT + 128);
  count_deg<<<gCnt, B256, 0, stream>>>(ei_u2u + NE, dinv_u, NE);   // in-deg(dst)
  count_deg<<<gCnt, B256, 0, stream>>>(ei_t2t + NE, dinv_t, NE);
  count_deg<<<gCnt, B256, 0, stream>>>(ei_u2t,      dsi_u2t, NE);  // out-deg(src)
  count_deg<<<gCnt, B256, 0, stream>>>(ei_u2t + NE, ddi_u2t, NE);  // in-deg(dst)
  count_deg<<<gCnt, B256, 0, stream>>>(ei_t2u,      dsi_t2u, NE);
  count_deg<<<gCnt, B256, 0, stream>>>(ei_t2u + NE, ddi_t2u, NE);
  deg_inv_self<<<gBig, B256, 0, stream>>>(dinv_u, NU);
  deg_inv_self<<<gBig, B256, 0, stream>>>(dinv_t, NT);
  deg_inv_bip<<<gBig, B256, 0, stream>>>(dsi_u2t, 2 * NU + 2 * NT);  // contiguous 4 arrays

  // --- embeddings (WMMA GEMMs) ---
  gemm_n64<<<gemmBlkU, B256, 0, stream>>>(x_user, W_emb_u, b_emb_u, hu, NU, 32);
  gemm_n64<<<gemmBlkT, B256, 0, stream>>>(x_txn,  W_emb_t, b_emb_t, ht, NT, 64);

  // --- GNN layers ---
  for (int l = 0; l < NLAYERS; ++l) {
    const float* W0   = convW + ((size_t)l * 4 + 0) * HD * HD;  // u2u
    const float* Wt   = convW + ((size_t)l * 4 + 1) * HD * HD;  // t2t
    const float* Wu2t = convW + ((size_t)l * 4 + 2) * HD * HD;
    const float* Wt2u = convW + ((size_t)l * 4 + 3) * HD * HD;
    const float* B0   = convB + ((size_t)l * 4 + 0) * HD;
    const float* Bt   = convB + ((size_t)l * 4 + 1) * HD;
    const float* Bu2t = convB + ((size_t)l * 4 + 2) * HD;
    const float* Bt2u = convB + ((size_t)l * 4 + 3) * HD;

    fill_f32<<<gBig, B256, 0, stream>>>(acc_u, 0.f, (long)(NU + NT) * HD);

    // u -> u
    gemm_n64<<<gemmBlkU, B256, 0, stream>>>(hu, W0, nullptr, m_a, NU, HD);
    add_self<<<gBig, B256, 0, stream>>>(acc_u, m_a, dinv_u, (long)NU * HD);
    scatter_edges<<<gBig, B256, 0, stream>>>(m_a, acc_u, ei_u2u, ei_u2u + NE,
                                             dinv_u, dinv_u, NE);
    // u -> t (reuses m_a; prior scatter completes first in stream order)
    gemm_n64<<<gemmBlkU, B256, 0, stream>>>(hu, Wu2t, nullptr, m_a, NU, HD);
    scatter_edges<<<gBig, B256, 0, stream>>>(m_a, acc_t, ei_u2t, ei_u2t + NE,
                                             dsi_u2t, ddi_u2t, NE);
    // t -> t
    gemm_n64<<<gemmBlkT, B256, 0, stream>>>(ht, Wt, nullptr, m_b, NT, HD);
    add_self<<<gBig, B256, 0, stream>>>(acc_t, m_b, dinv_t, (long)NT * HD);
    scatter_edges<<<gBig, B256, 0, stream>>>(m_b, acc_t, ei_t2t, ei_t2t + NE,
                                             dinv_t, dinv_t, NE);
    // t -> u
    gemm_n64<<<gemmBlkT, B256, 0, stream>>>(ht, Wt2u, nullptr, m_b, NT, HD);
    scatter_edges<<<gBig, B256, 0, stream>>>(m_b, acc_u, ei_t2u, ei_t2u + NE,
                                             dsi_t2u, ddi_t2u, NE);

    // mean across relations + bias + ReLU (in place)
    combine_relu<<<gBig, B256, 0, stream>>>(hu, acc_u, B0, Bt2u, (long)NU * HD);
    combine_relu<<<gBig, B256, 0, stream>>>(ht, acc_t, Bt, Bu2t, (long)NT * HD);
  }

  // --- readout + MLP head ---
  colsum<<<512, B256, 0, stream>>>(hu, gsum, NU);
  colsum<<<512, B256, 0, stream>>>(ht, gsum + 64, NT);
  final_mlp<<<1, 64, 0, stream>>>(gsum, gsum + 64, W1, b1, W2, b2, out);
}